// AttnDecoderRNN_69750268887159
// MI455X (gfx1250) — compile-verified
//
#include <hip/hip_runtime.h>
#include <hip/hip_bf16.h>
#include <math.h>

// Problem dims (match reference)
#define BB  512
#define SS  512
#define HH  128
#define TT  8
#define OUTF 136

typedef __attribute__((ext_vector_type(2))) float v2f;
typedef __attribute__((ext_vector_type(8))) float v8f;

// ---------------------------------------------------------------------------
// init: h0 = mean(encoder_hidden, axis=0); emb = tpo @ emb_W^T + emb_b
// one thread per (b, h)
// ---------------------------------------------------------------------------
__global__ void init_kernel(const float* __restrict__ enc_hid,   // [2,B,H]
                            const float* __restrict__ tpo,       // [B,1,OUT]
                            const float* __restrict__ emb_W,     // [H,OUT]
                            const float* __restrict__ emb_b,     // [H]
                            float* __restrict__ h,               // [B,H]
                            float* __restrict__ emb)             // [B,H]
{
    int tid = blockIdx.x * blockDim.x + threadIdx.x;
    int b = tid / HH;
    int hh = tid % HH;
    h[tid] = 0.5f * (enc_hid[(size_t)b * HH + hh] +
                     enc_hid[(size_t)BB * HH + (size_t)b * HH + hh]);
    float acc = emb_b[hh];
    const float* wrow = emb_W + (size_t)hh * OUTF;
    const float* xv = tpo + (size_t)b * OUTF;
    #pragma unroll 8
    for (int o = 0; o < OUTF; ++o) acc += wrow[o] * xv[o];
    emb[tid] = acc;
}

// ---------------------------------------------------------------------------
// q = h @ Wa^T + Wa_b   (one thread per (b,h))
// ---------------------------------------------------------------------------
__global__ void q_kernel(const float* __restrict__ h,    // [B,H]
                         const float* __restrict__ Wa,   // [H,H]
                         const float* __restrict__ Wa_b, // [H]
                         float* __restrict__ q)          // [B,H]
{
    int tid = blockIdx.x * blockDim.x + threadIdx.x;
    int b = tid / HH;
    int hh = tid % HH;
    float acc = Wa_b[hh];
    const float* wrow = Wa + (size_t)hh * HH;
    const float* hv = h + (size_t)b * HH;
    #pragma unroll 8
    for (int k = 0; k < HH; ++k) acc += wrow[k] * hv[k];
    q[tid] = acc;
}

// ---------------------------------------------------------------------------
// Fused Bahdanau scores with f32 WMMA:
//   scores[b,s] = Va . tanh( (keys[b,s,:] @ Ua^T + Ua_b) + q[b,:] ) + Va_b
// One wave per (b, 16-row s-tile). The 16x16 proj tiles are produced with
// V_WMMA_F32_16X16X4_F32 (K=128 -> 32 WMMA per tile, 8 h-tiles per wave).
// A-rows (keys) cached in VGPRs; tanh/Va reduce fused on accumulators.
// ---------------------------------------------------------------------------
__global__ void scores_kernel(const float* __restrict__ keys,  // [B,S,H]
                              const float* __restrict__ Ua,    // [H,H]
                              const float* __restrict__ Ua_b,  // [H]
                              const float* __restrict__ q,     // [B,H]
                              const float* __restrict__ Va,    // [H]
                              const float* __restrict__ Va_b,  // [1]
                              float* __restrict__ scores)      // [B,S]
{
    const int lane  = threadIdx.x & 31;
    const int wid   = threadIdx.x >> 5;
    const int gwave = blockIdx.x * (blockDim.x >> 5) + wid;

    const int num_stiles = SS / 16;            // 32
    const int b  = gwave / num_stiles;
    const int s0 = (gwave % num_stiles) * 16;

    const int lhalf = lane >> 4;               // 0 or 1
    const int l16   = lane & 15;

    // A-matrix: lane holds keys row (s0 + l16), columns k = 4*kk + 2*lhalf + {0,1}
    const float* keyrow = keys + ((size_t)b * SS + (size_t)(s0 + l16)) * HH + 2 * lhalf;
    v2f areg[32];
    #pragma unroll
    for (int kk = 0; kk < 32; ++kk)
        areg[kk] = *(const v2f*)(keyrow + 4 * kk);

    float acc[8];
    #pragma unroll
    for (int j = 0; j < 8; ++j) acc[j] = 0.0f;

    const float* qrow = q + (size_t)b * HH;

    for (int ht = 0; ht < 8; ++ht) {
        const int hbase = ht * 16;
        // B-matrix: B[K][N] = Ua[hbase + N][k]; lane holds row (hbase+l16),
        // columns k = 4*kk + 2*lhalf + {0,1}
        const float* uarow = Ua + (size_t)(hbase + l16) * HH + 2 * lhalf;
        v8f c = {};
        #pragma unroll
        for (int kk = 0; kk < 32; ++kk) {
            v2f bm = *(const v2f*)(uarow + 4 * kk);
            c = __builtin_amdgcn_wmma_f32_16x16x4_f32(
                    false, areg[kk], false, bm, (short)0, c, false, false);
        }
        // lane's column of D is h_out = hbase + l16
        const float qv  = qrow[hbase + l16] + Ua_b[hbase + l16];
        const float vav = Va[hbase + l16];
        #pragma unroll
        for (int j = 0; j < 8; ++j)
            acc[j] += vav * tanhf(c[j] + qv);
    }

    // Reduce over the 16 lanes of each half (the N/h_out dimension).
    #pragma unroll
    for (int j = 0; j < 8; ++j) {
        float v = acc[j];
        v += __shfl_xor(v, 1, 32);
        v += __shfl_xor(v, 2, 32);
        v += __shfl_xor(v, 4, 32);
        v += __shfl_xor(v, 8, 32);
        acc[j] = v;
    }

    const float vab = Va_b[0];
    // D row for VGPR j is s_local = j + 8*lhalf; lane l16==j writes it.
    #pragma unroll
    for (int j = 0; j < 8; ++j)
        if (l16 == j)
            scores[(size_t)b * SS + (s0 + j + 8 * lhalf)] = acc[j] + vab;
}

// ---------------------------------------------------------------------------
// softmax over S per batch row; also writes attention output slice for step t
// ---------------------------------------------------------------------------
__global__ void softmax_kernel(const float* __restrict__ scores, // [B,S]
                               float* __restrict__ w,            // [B,S]
                               float* __restrict__ attn,         // [B,T,S]
                               int t)
{
    const int b = blockIdx.x;
    const int tid = threadIdx.x;
    __shared__ float red[256];

    float m = -3.4e38f;
    for (int s = tid; s < SS; s += 256)
        m = fmaxf(m, scores[(size_t)b * SS + s]);
    red[tid] = m;
    __syncthreads();
    for (int off = 128; off > 0; off >>= 1) {
        if (tid < off) red[tid] = fmaxf(red[tid], red[tid + off]);
        __syncthreads();
    }
    m = red[0];
    __syncthreads();

    float sum = 0.0f;
    for (int s = tid; s < SS; s += 256) {
        float e = __expf(scores[(size_t)b * SS + s] - m);
        w[(size_t)b * SS + s] = e;
        sum += e;
    }
    red[tid] = sum;
    __syncthreads();
    for (int off = 128; off > 0; off >>= 1) {
        if (tid < off) red[tid] += red[tid + off];
        __syncthreads();
    }
    const float inv = 1.0f / red[0];

    for (int s = tid; s < SS; s += 256) {
        float wv = w[(size_t)b * SS + s] * inv;
        w[(size_t)b * SS + s] = wv;
        attn[((size_t)b * TT + t) * SS + s] = wv;
    }
}

// ---------------------------------------------------------------------------
// context[b,h] = sum_s w[b,s] * keys[b,s,h]   (block per b, 128 threads)
// ---------------------------------------------------------------------------
__global__ void context_kernel(const float* __restrict__ keys, // [B,S,H]
                               const float* __restrict__ w,    // [B,S]
                               float* __restrict__ ctx)        // [B,H]
{
    const int b = blockIdx.x;
    const int hh = threadIdx.x;
    __shared__ float wsh[128];
    float acc = 0.0f;
    for (int sc = 0; sc < SS; sc += 128) {
        wsh[hh] = w[(size_t)b * SS + sc + hh];
        __syncthreads();
        const float* krow = keys + ((size_t)b * SS + sc) * HH + hh;
        #pragma unroll 8
        for (int i = 0; i < 128; ++i)
            acc += wsh[i] * krow[(size_t)i * HH];
        __syncthreads();
    }
    ctx[(size_t)b * HH + hh] = acc;
}

// ---------------------------------------------------------------------------
// Fused GRU cell + output projection (block per b, 384 threads)
// gate layout (PyTorch): r, z, n
// ---------------------------------------------------------------------------
__global__ void gru_out_kernel(const float* __restrict__ emb,   // [B,H]
                               const float* __restrict__ ctx,   // [B,H]
                               float* __restrict__ h,           // [B,H] (in/out)
                               const float* __restrict__ W_ih,  // [3H,2H]
                               const float* __restrict__ W_hh,  // [3H,H]
                               const float* __restrict__ b_ih,  // [3H]
                               const float* __restrict__ b_hh,  // [3H]
                               const float* __restrict__ out_W, // [OUT,H]
                               const float* __restrict__ out_b, // [OUT]
                               float* __restrict__ dec_out,     // [B,T,OUT]
                               int t)
{
    const int b = blockIdx.x;
    const int tid = threadIdx.x;

    __shared__ float xsh[2 * HH];   // [emb | ctx]
    __shared__ float hsh[HH];
    __shared__ float gish[3 * HH];
    __shared__ float ghsh[3 * HH];
    __shared__ float hnsh[HH];

    if (tid < HH) {
        xsh[tid] = emb[(size_t)b * HH + tid];
        hsh[tid] = h[(size_t)b * HH + tid];
    } else if (tid < 2 * HH) {
        xsh[tid] = ctx[(size_t)b * HH + (tid - HH)];
    }
    __syncthreads();

    // gi = W_ih @ x + b_ih ; gh = W_hh @ h + b_hh   (tid in [0,384))
    {
        float gi = b_ih[tid];
        const float* wi = W_ih + (size_t)tid * (2 * HH);
        #pragma unroll 8
        for (int k = 0; k < 2 * HH; ++k) gi += wi[k] * xsh[k];
        float gh = b_hh[tid];
        const float* wh = W_hh + (size_t)tid * HH;
        #pragma unroll 8
        for (int k = 0; k < HH; ++k) gh += wh[k] * hsh[k];
        gish[tid] = gi;
        ghsh[tid] = gh;
    }
    __syncthreads();

    if (tid < HH) {
        float r = 1.0f / (1.0f + __expf(-(gish[tid] + ghsh[tid])));
        float z = 1.0f / (1.0f + __expf(-(gish[HH + tid] + ghsh[HH + tid])));
        float n = tanhf(gish[2 * HH + tid] + r * ghsh[2 * HH + tid]);
        float hn = (1.0f - z) * n + z * hsh[tid];
        hnsh[tid] = hn;
        h[(size_t)b * HH + tid] = hn;
    }
    __syncthreads();

    if (tid < OUTF) {
        float acc = out_b[tid];
        const float* wrow = out_W + (size_t)tid * HH;
        #pragma unroll 8
        for (int k = 0; k < HH; ++k) acc += wrow[k] * hnsh[k];
        dec_out[((size_t)b * TT + t) * OUTF + tid] = acc;
    }
}

// ---------------------------------------------------------------------------
extern "C" void kernel_launch(void* const* d_in, const int* in_sizes, int n_in,
                              void* d_out, int out_size, void* d_ws, size_t ws_size,
                              hipStream_t stream) {
    (void)in_sizes; (void)n_in; (void)out_size; (void)ws_size;

    const float* enc_out = (const float*)d_in[0];   // [B,S,H]
    const float* enc_hid = (const float*)d_in[1];   // [2,B,H]
    const float* tpo     = (const float*)d_in[2];   // [B,1,OUT]
    /* d_in[3] = target_tensor (unused under teacher forcing fixed input) */
    const float* emb_W   = (const float*)d_in[4];   // [H,OUT]
    const float* emb_b   = (const float*)d_in[5];   // [H]
    const float* Wa      = (const float*)d_in[6];   // [H,H]
    const float* Wa_b    = (const float*)d_in[7];   // [H]
    const float* Ua      = (const float*)d_in[8];   // [H,H]
    const float* Ua_b    = (const float*)d_in[9];   // [H]
    const float* Va      = (const float*)d_in[10];  // [1,H]
    const float* Va_b    = (const float*)d_in[11];  // [1]
    const float* W_ih    = (const float*)d_in[12];  // [3H,2H]
    const float* W_hh    = (const float*)d_in[13];  // [3H,H]
    const float* b_ih    = (const float*)d_in[14];  // [3H]
    const float* b_hh    = (const float*)d_in[15];  // [3H]
    const float* out_W   = (const float*)d_in[16];  // [OUT,H]
    const float* out_b   = (const float*)d_in[17];  // [OUT]

    float* out     = (float*)d_out;
    float* dec_out = out;                                   // B*T*OUT
    float* dec_hid = out + (size_t)BB * TT * OUTF;          // B*H
    float* attn    = dec_hid + (size_t)BB * HH;             // B*T*S

    float* ws   = (float*)d_ws;
    float* h    = ws;                          // B*H
    float* emb  = h   + (size_t)BB * HH;       // B*H
    float* q    = emb + (size_t)BB * HH;       // B*H
    float* sc   = q   + (size_t)BB * HH;       // B*S
    float* w    = sc  + (size_t)BB * SS;       // B*S
    float* ctx  = w   + (size_t)BB * SS;       // B*H

    init_kernel<<<(BB * HH) / 256, 256, 0, stream>>>(enc_hid, tpo, emb_W, emb_b, h, emb);

    const int score_blocks = (BB * (SS / 16)) / 8;   // 8 waves/block, 1 wave per tile
    for (int t = 0; t < TT; ++t) {
        q_kernel<<<(BB * HH) / 256, 256, 0, stream>>>(h, Wa, Wa_b, q);
        scores_kernel<<<score_blocks, 256, 0, stream>>>(enc_out, Ua, Ua_b, q, Va, Va_b, sc);
        softmax_kernel<<<BB, 256, 0, stream>>>(sc, w, attn, t);
        context_kernel<<<BB, 128, 0, stream>>>(enc_out, w, ctx);
        gru_out_kernel<<<BB, 384, 0, stream>>>(emb, ctx, h, W_ih, W_hh, b_ih, b_hh,
                                               out_W, out_b, dec_out, t);
    }

    hipMemcpyAsync(dec_hid, h, (size_t)BB * HH * sizeof(float),
                   hipMemcpyDeviceToDevice, stream);
}